// PIT_LFMMILoss_66202625900735
// MI455X (gfx1250) — compile-verified
//
#include <hip/hip_runtime.h>
#include <hip/hip_bf16.h>
#include <stdint.h>

// ---------------------------------------------------------------------------
// PIT-LFMMI loss for MI455X (gfx1250, wave32).
//   A: denominator forward algorithm, 1 block per (speaker,batch) sequence.
//   B: numerator forward algorithm,  1 block per (llh-speaker, graph, batch).
//   C: permutation scoring via exact V_WMMA_F32_16X16X4_F32 + final reduce.
// CDNA5 paths: global_load_async_to_lds_b128 + s_wait_asynccnt double buffer
// of the per-step emission row, LDS float atomics for segment-LSE, LDS-resident
// alpha, b128-vectorized arc fetch (4 arcs/thread), one f32 WMMA.
// ---------------------------------------------------------------------------

#define NEGF (-1e30f)

static __device__ __forceinline__ void async_copy16(uint32_t lds_off, const void* gaddr) {
  unsigned long long ga = (unsigned long long)(uintptr_t)gaddr;
  asm volatile("global_load_async_to_lds_b128 %0, %1, off"
               :: "v"(lds_off), "v"(ga) : "memory");
}
static __device__ __forceinline__ void wait_async0() {
  asm volatile("s_wait_asynccnt 0" ::: "memory");
}
static __device__ __forceinline__ uint32_t lds_off32(const void* p) {
  // generic pointer to LDS: low 32 bits are the workgroup-relative LDS offset
  return (uint32_t)(uintptr_t)p;
}
// float atomic-max on LDS via signed/unsigned integer ordering trick
static __device__ __forceinline__ void atomicMaxFloatLds(float* a, float v) {
  if (v >= 0.0f) atomicMax((int*)a, __float_as_int(v));
  else           atomicMin((unsigned int*)a, (unsigned int)__float_as_int(v));
}

// ---------------------------------------------------------------------------
// Kernel A: denominator forward.  grid = 32 (= S*B), block = 512 (16 waves)
// ---------------------------------------------------------------------------
__global__ __launch_bounds__(512)
void den_forward(const float* __restrict__ est, const int* __restrict__ seqlen,
                 const int* __restrict__ src, const int* __restrict__ dst,
                 const int* __restrict__ lab, const float* __restrict__ wgt,
                 const float* __restrict__ start, const float* __restrict__ fin,
                 float* __restrict__ den_out)
{
  const int B = 16, T = 300, C = 3000;
  const int NS = 3000, NA = 60000;

  __shared__ __align__(16) float alpha[3000];
  __shared__ __align__(16) float mbuf[3000];
  __shared__ __align__(16) float sbuf[3000];
  __shared__ __align__(16) float llh[2][3008];
  __shared__ float red[2];

  const int blk  = blockIdx.x;
  const int s    = blk / B;
  const int b    = blk % B;
  const int tid  = threadIdx.x;
  const int nthr = blockDim.x;
  const int L    = seqlen[b];

  const float* row0 = est + ((size_t)(s * B + b) * T) * C;

  const int4*   src4 = (const int4*)src;     // 16B-aligned, NA % 4 == 0
  const int4*   dst4 = (const int4*)dst;
  const int4*   lab4 = (const int4*)lab;
  const float4* wgt4 = (const float4*)wgt;
  const float4* st4  = (const float4*)start;

  float4* alpha4 = (float4*)alpha;
  float4* mbuf4  = (float4*)mbuf;
  float4* sbuf4  = (float4*)sbuf;

  const float MINF = __int_as_float(0xff800000u);   // -inf

  for (int j = tid; j < NS / 4; j += nthr) alpha4[j] = st4[j];

  // prologue: async load emission row t=0 into llh[0]  (750 x 16B chunks)
  for (int i = tid; i < C / 4; i += nthr)
    async_copy16(lds_off32(&llh[0][i * 4]), (const void*)(row0 + i * 4));

  for (int t = 0; t < L; ++t) {
    const int buf = t & 1;
    wait_async0();          // this wave's llh[buf] chunks are in LDS
    __syncthreads();        // all waves' chunks are in LDS

    // prefetch t+1 emission row into the other buffer
    if (t + 1 < T) {
      const float* g = row0 + (size_t)(t + 1) * C;
      const int nb = buf ^ 1;
      for (int i = tid; i < C / 4; i += nthr)
        async_copy16(lds_off32(&llh[nb][i * 4]), (const void*)(g + i * 4));
    }

    const float4 minf4 = make_float4(MINF, MINF, MINF, MINF);
    const float4 zero4 = make_float4(0.f, 0.f, 0.f, 0.f);
    for (int j = tid; j < NS / 4; j += nthr) { mbuf4[j] = minf4; sbuf4[j] = zero4; }
    __syncthreads();

    // pass 1: per-destination max of alpha[src] + w + llh[lab]   (4 arcs/thread)
    for (int j = tid; j < NA / 4; j += nthr) {
      int4 s4 = src4[j]; int4 d4 = dst4[j]; int4 l4 = lab4[j]; float4 w4 = wgt4[j];
      atomicMaxFloatLds(&mbuf[d4.x], alpha[s4.x] + w4.x + llh[buf][l4.x]);
      atomicMaxFloatLds(&mbuf[d4.y], alpha[s4.y] + w4.y + llh[buf][l4.y]);
      atomicMaxFloatLds(&mbuf[d4.z], alpha[s4.z] + w4.z + llh[buf][l4.z]);
      atomicMaxFloatLds(&mbuf[d4.w], alpha[s4.w] + w4.w + llh[buf][l4.w]);
    }
    __syncthreads();

    // pass 2: per-destination sum of exp(score - max)
    for (int j = tid; j < NA / 4; j += nthr) {
      int4 s4 = src4[j]; int4 d4 = dst4[j]; int4 l4 = lab4[j]; float4 w4 = wgt4[j];
      atomicAdd(&sbuf[d4.x], __expf(alpha[s4.x] + w4.x + llh[buf][l4.x] - mbuf[d4.x]));
      atomicAdd(&sbuf[d4.y], __expf(alpha[s4.y] + w4.y + llh[buf][l4.y] - mbuf[d4.y]));
      atomicAdd(&sbuf[d4.z], __expf(alpha[s4.z] + w4.z + llh[buf][l4.z] - mbuf[d4.z]));
      atomicAdd(&sbuf[d4.w], __expf(alpha[s4.w] + w4.w + llh[buf][l4.w] - mbuf[d4.w]));
    }
    __syncthreads();

    for (int j = tid; j < NS / 4; j += nthr) {
      float4 sv = sbuf4[j], mv = mbuf4[j], av;
      av.x = (sv.x > 0.f) ? (mv.x + __logf(sv.x)) : NEGF;
      av.y = (sv.y > 0.f) ? (mv.y + __logf(sv.y)) : NEGF;
      av.z = (sv.z > 0.f) ? (mv.z + __logf(sv.z)) : NEGF;
      av.w = (sv.w > 0.f) ? (mv.w + __logf(sv.w)) : NEGF;
      alpha4[j] = av;
    }
    __syncthreads();
  }
  wait_async0();  // drain unused prefetch

  // final logZ = LSE(alpha + final)
  if (tid == 0) { red[0] = MINF; red[1] = 0.0f; }
  __syncthreads();
  float lm = -3.0e38f;
  for (int st = tid; st < NS; st += nthr) lm = fmaxf(lm, alpha[st] + fin[st]);
  atomicMaxFloatLds(&red[0], lm);
  __syncthreads();
  const float M = red[0];
  float le = 0.0f;
  for (int st = tid; st < NS; st += nthr) le += __expf(alpha[st] + fin[st] - M);
  atomicAdd(&red[1], le);
  __syncthreads();
  if (tid == 0) den_out[s * B + b] = M + __logf(red[1]);
}

// ---------------------------------------------------------------------------
// Kernel B: numerator forward for every (llh-speaker, graph) combo.
// grid = 64 (= S * P * B), block = 256.  Output NL[s*2+g][b].
// ---------------------------------------------------------------------------
__global__ __launch_bounds__(256)
void num_forward(const float* __restrict__ est, const int* __restrict__ seqlen,
                 const int* __restrict__ src, const int* __restrict__ dst,
                 const int* __restrict__ lab, const float* __restrict__ wgt,
                 const float* __restrict__ start, const float* __restrict__ fin,
                 float* __restrict__ nl_out)
{
  const int B = 16, T = 300, C = 3000;
  const int NS = 400, NA = 1200;

  __shared__ __align__(16) float alpha[400];
  __shared__ __align__(16) float mbuf[400];
  __shared__ __align__(16) float sbuf[400];
  __shared__ __align__(16) float llh[2][3008];
  __shared__ float red[2];

  const int blk  = blockIdx.x;
  const int b    = blk % B;
  const int g    = (blk / B) & 1;       // numerator graph index
  const int sll  = blk / (B * 2);       // which speaker's llh stream
  const int tid  = threadIdx.x;
  const int nthr = blockDim.x;
  const int L    = seqlen[b];

  const float* row0 = est + ((size_t)(sll * B + b) * T) * C;
  const int   gb    = g * B + b;
  // per-(graph,batch) slices; offsets are 16B-aligned (1200*4, 400*4 bytes)
  const int4*   src4 = (const int4*)  (src   + (size_t)gb * NA);
  const int4*   dst4 = (const int4*)  (dst   + (size_t)gb * NA);
  const int4*   lab4 = (const int4*)  (lab   + (size_t)gb * NA);
  const float4* wgt4 = (const float4*)(wgt   + (size_t)gb * NA);
  const float4* st4  = (const float4*)(start + (size_t)gb * NS);
  const float*  gfn  = fin + (size_t)gb * NS;

  float4* alpha4 = (float4*)alpha;
  float4* mbuf4  = (float4*)mbuf;
  float4* sbuf4  = (float4*)sbuf;

  const float MINF = __int_as_float(0xff800000u);

  for (int j = tid; j < NS / 4; j += nthr) alpha4[j] = st4[j];

  for (int i = tid; i < C / 4; i += nthr)
    async_copy16(lds_off32(&llh[0][i * 4]), (const void*)(row0 + i * 4));

  for (int t = 0; t < L; ++t) {
    const int buf = t & 1;
    wait_async0();
    __syncthreads();

    if (t + 1 < T) {
      const float* gp = row0 + (size_t)(t + 1) * C;
      const int nb = buf ^ 1;
      for (int i = tid; i < C / 4; i += nthr)
        async_copy16(lds_off32(&llh[nb][i * 4]), (const void*)(gp + i * 4));
    }

    const float4 minf4 = make_float4(MINF, MINF, MINF, MINF);
    const float4 zero4 = make_float4(0.f, 0.f, 0.f, 0.f);
    for (int j = tid; j < NS / 4; j += nthr) { mbuf4[j] = minf4; sbuf4[j] = zero4; }
    __syncthreads();

    for (int j = tid; j < NA / 4; j += nthr) {
      int4 s4 = src4[j]; int4 d4 = dst4[j]; int4 l4 = lab4[j]; float4 w4 = wgt4[j];
      atomicMaxFloatLds(&mbuf[d4.x], alpha[s4.x] + w4.x + llh[buf][l4.x]);
      atomicMaxFloatLds(&mbuf[d4.y], alpha[s4.y] + w4.y + llh[buf][l4.y]);
      atomicMaxFloatLds(&mbuf[d4.z], alpha[s4.z] + w4.z + llh[buf][l4.z]);
      atomicMaxFloatLds(&mbuf[d4.w], alpha[s4.w] + w4.w + llh[buf][l4.w]);
    }
    __syncthreads();

    for (int j = tid; j < NA / 4; j += nthr) {
      int4 s4 = src4[j]; int4 d4 = dst4[j]; int4 l4 = lab4[j]; float4 w4 = wgt4[j];
      atomicAdd(&sbuf[d4.x], __expf(alpha[s4.x] + w4.x + llh[buf][l4.x] - mbuf[d4.x]));
      atomicAdd(&sbuf[d4.y], __expf(alpha[s4.y] + w4.y + llh[buf][l4.y] - mbuf[d4.y]));
      atomicAdd(&sbuf[d4.z], __expf(alpha[s4.z] + w4.z + llh[buf][l4.z] - mbuf[d4.z]));
      atomicAdd(&sbuf[d4.w], __expf(alpha[s4.w] + w4.w + llh[buf][l4.w] - mbuf[d4.w]));
    }
    __syncthreads();

    for (int j = tid; j < NS / 4; j += nthr) {
      float4 sv = sbuf4[j], mv = mbuf4[j], av;
      av.x = (sv.x > 0.f) ? (mv.x + __logf(sv.x)) : NEGF;
      av.y = (sv.y > 0.f) ? (mv.y + __logf(sv.y)) : NEGF;
      av.z = (sv.z > 0.f) ? (mv.z + __logf(sv.z)) : NEGF;
      av.w = (sv.w > 0.f) ? (mv.w + __logf(sv.w)) : NEGF;
      alpha4[j] = av;
    }
    __syncthreads();
  }
  wait_async0();

  if (tid == 0) { red[0] = MINF; red[1] = 0.0f; }
  __syncthreads();
  float lm = -3.0e38f;
  for (int st = tid; st < NS; st += nthr) lm = fmaxf(lm, alpha[st] + gfn[st]);
  atomicMaxFloatLds(&red[0], lm);
  __syncthreads();
  const float M = red[0];
  float le = 0.0f;
  for (int st = tid; st < NS; st += nthr) le += __expf(alpha[st] + gfn[st] - M);
  atomicAdd(&red[1], le);
  __syncthreads();
  if (tid == 0) nl_out[(sll * 2 + g) * B + b] = M + __logf(red[1]);
}

// ---------------------------------------------------------------------------
// Kernel C: permutation scoring as an exact 16x16x4 F32 WMMA + final loss.
//   perm_sum[p][b] = sum_k Pmat[p][k] * NL[k][b],  k = s*2 + g,
//   Pmat[0] = {1,0,0,1} (identity perm), Pmat[1] = {0,1,1,0} (swap).
//   loss = sum_b [ (den[0][b] + den[1][b]) - max_p perm_sum[p][b] ].
// Single wave32, EXEC all ones (WMMA requirement).
// ---------------------------------------------------------------------------
typedef float v2f __attribute__((ext_vector_type(2)));
typedef float v8f __attribute__((ext_vector_type(8)));

__global__ __launch_bounds__(32)
void pit_combine(const float* __restrict__ den_ws, const float* __restrict__ nl_ws,
                 float* __restrict__ out)
{
  const int lane = threadIdx.x;
  const int M  = lane & 15;
  const int hi = lane >> 4;   // 0: K=0/1 half, 1: K=2/3 half

  // A matrix 16x4 (rows = permutations, cols = (speaker,graph) combos)
  v2f A;
  A.x = (M == (hi ? 1 : 0)) ? 1.0f : 0.0f;   // Pmat[M][hi?2:0]
  A.y = (M == (hi ? 0 : 1)) ? 1.0f : 0.0f;   // Pmat[M][hi?3:1]

  // B matrix 4x16 (rows = combos, cols = batch)
  const int N  = lane & 15;
  const int k0 = hi ? 2 : 0;
  v2f Bm;
  Bm.x = nl_ws[(k0 + 0) * 16 + N];
  Bm.y = nl_ws[(k0 + 1) * 16 + N];

  v8f acc = {};
  acc = __builtin_amdgcn_wmma_f32_16x16x4_f32(
      /*neg_a=*/false, A, /*neg_b=*/false, Bm,
      /*c_mod=*/(short)0, acc, /*reuse_a=*/false, /*reuse_b=*/false);
  // D layout: VGPR r, lanes 0-15 -> row M=r, col N=lane.
  // acc[0] = perm0 sum for batch=lane, acc[1] = perm1 sum (lanes 0-15).

  __shared__ float part[16];
  if (lane < 16) {
    float best = fmaxf(acc[0], acc[1]);               // argmax perm -> its sum
    float den  = den_ws[0 * 16 + lane] + den_ws[1 * 16 + lane];
    part[lane] = den - best;                          // -(num - den) summed over s
  }
  __syncthreads();
  if (lane == 0) {
    float loss = 0.0f;
    for (int i = 0; i < 16; ++i) loss += part[i];
    out[0] = loss;
  }
}

// ---------------------------------------------------------------------------
extern "C" void kernel_launch(void* const* d_in, const int* in_sizes, int n_in,
                              void* d_out, int out_size, void* d_ws, size_t ws_size,
                              hipStream_t stream) {
  (void)in_sizes; (void)n_in; (void)out_size; (void)ws_size;

  const float* est     = (const float*)d_in[0];
  const int*   seqlen  = (const int*)  d_in[1];
  const int*   nsrc    = (const int*)  d_in[2];
  const int*   ndst    = (const int*)  d_in[3];
  const int*   nlab    = (const int*)  d_in[4];
  const float* nw      = (const float*)d_in[5];
  const float* nstart  = (const float*)d_in[6];
  const float* nfinal  = (const float*)d_in[7];
  const int*   dsrc    = (const int*)  d_in[8];
  const int*   ddst    = (const int*)  d_in[9];
  const int*   dlab    = (const int*)  d_in[10];
  const float* dw      = (const float*)d_in[11];
  const float* dstart  = (const float*)d_in[12];
  const float* dfinal  = (const float*)d_in[13];

  float* ws     = (float*)d_ws;
  float* den_ws = ws;        // 32 floats: den logZ [s][b]
  float* nl_ws  = ws + 32;   // 64 floats: num logZ [s*2+g][b]

  den_forward<<<32, 512, 0, stream>>>(est, seqlen, dsrc, ddst, dlab, dw,
                                      dstart, dfinal, den_ws);
  num_forward<<<64, 256, 0, stream>>>(est, seqlen, nsrc, ndst, nlab, nw,
                                      nstart, nfinal, nl_ws);
  pit_combine<<<1, 32, 0, stream>>>(den_ws, nl_ws, (float*)d_out);
}